// LoRALinear_14448269984184
// MI455X (gfx1250) — compile-verified
//
#include <hip/hip_runtime.h>

// LoRA linear for MI455X (gfx1250, wave32):
//   t   = 2 * (x @ A^T)          : kernel 1, read-bound  (134 MB x stream)
//   out = t @ B^T + bias         : kernel 2, store-bound (134 MB out stream), f32 WMMA
//
// x: [8192, 4096] f32, A: [16, 4096], B: [4096, 16], bias: [4096]
// t workspace: 8192*16 f32 = 512 KB in d_ws.

#define IN_F   4096
#define OUT_F  4096
#define RANK   16
#define NTOK   8192          // 4 * 2048
#define TOK_PER_WG 16

typedef float v2f __attribute__((ext_vector_type(2)));
typedef float v8f __attribute__((ext_vector_type(8)));

// ---------------------------------------------------------------------------
// Kernel 1: t[tok][r] = 2 * sum_k x[tok][k] * A[r][k]
// 256 threads = 8 waves; each wave owns 2 tokens; lanes split K with float4
// loads (lane-consecutive 16B -> 512 B fully-coalesced per instruction).
// A is streamed identically by all 8 waves -> WGP$ hit, ~256 KB L2/WG.
// ---------------------------------------------------------------------------
__global__ __launch_bounds__(256) void lora_xat(const float* __restrict__ x,
                                                const float* __restrict__ A,
                                                float* __restrict__ t) {
  __shared__ float red[TOK_PER_WG][RANK][32];   // 32 KB lane-partial scratch

  const int lane    = threadIdx.x & 31;
  const int wave    = threadIdx.x >> 5;
  const int tokBase = blockIdx.x * TOK_PER_WG;
  const int tok0    = tokBase + wave * 2;

  const float4* __restrict__ xr0 = (const float4*)(x + (size_t)tok0 * IN_F);
  const float4* __restrict__ xr1 = (const float4*)(x + (size_t)(tok0 + 1) * IN_F);

  float acc0[RANK], acc1[RANK];
#pragma unroll
  for (int r = 0; r < RANK; ++r) { acc0[r] = 0.0f; acc1[r] = 0.0f; }

  // 4096 floats / (32 lanes * 4 per float4) = 32 steps
  for (int j = 0; j < IN_F / 128; ++j) {
    const int kv = j * 32 + lane;               // float4 index within row
    const float4 x0 = xr0[kv];
    const float4 x1 = xr1[kv];
#pragma unroll
    for (int r = 0; r < RANK; ++r) {
      const float4 a = ((const float4*)(A + (size_t)r * IN_F))[kv];
      acc0[r] = fmaf(x0.x, a.x, fmaf(x0.y, a.y, fmaf(x0.z, a.z, fmaf(x0.w, a.w, acc0[r]))));
      acc1[r] = fmaf(x1.x, a.x, fmaf(x1.y, a.y, fmaf(x1.z, a.z, fmaf(x1.w, a.w, acc1[r]))));
    }
  }

#pragma unroll
  for (int r = 0; r < RANK; ++r) {
    red[wave * 2 + 0][r][lane] = acc0[r];
    red[wave * 2 + 1][r][lane] = acc1[r];
  }
  __syncthreads();

  // 256 threads: one (token, rank) each; sum 32 lane partials (float4 reads).
  const int tok = threadIdx.x >> 4;             // 0..15
  const int r   = threadIdx.x & 15;             // 0..15
  const float4* v = (const float4*)red[tok][r];
  float s = 0.0f;
#pragma unroll
  for (int i = 0; i < 8; ++i) { const float4 p = v[i]; s += (p.x + p.y) + (p.z + p.w); }

  t[(size_t)(tokBase + tok) * RANK + r] = 2.0f * s;   // fold LoRA scaling here
}

// ---------------------------------------------------------------------------
// Kernel 2: out[tok][n] = sum_r t[tok][r] * B[n][r] + bias[n]
// One wave -> 16 tokens x 64 columns via 16 x V_WMMA_F32_16X16X4_F32.
// Operand layouts (ISA 7.12.2, 32-bit, wave32):
//   A 16x4 : lane l, vgpr v -> M = l%16, K = v + 2*(l/16)  -> consecutive pair = b64 load
//   B 4x16 : lane l, vgpr v -> N = l%16, K = v + 2*(l/16)  -> consecutive pair = b64 load
//   D 16x16: lane l, vgpr v -> M = v + 8*(l/16), N = l%16
// Schedule: (1) ALL operand loads hoisted -> one load clause, one wait, then
// 16 WMMAs back-to-back; (2) s-outer / n-inner so consecutive WMMAs hit
// independent accumulators (4-way ILP, dependent link 4 insts away);
// (3) bias pre-splatted into C (every D vgpr of a lane shares N = l%16).
// Results are repacked through LDS so stores are b128 with 256 B contiguity.
// ---------------------------------------------------------------------------
#define NT 4                                    // 16-col tiles per wave (64 cols)

__global__ __launch_bounds__(256) void lora_tbt(const float* __restrict__ t,
                                                const float* __restrict__ Bw,
                                                const float* __restrict__ bias,
                                                float* __restrict__ out) {
  __shared__ float stile[8][16][68];            // 68-pad: conflict-free repack (~34 KB)

  const int lane = threadIdx.x & 31;
  const int wave = threadIdx.x >> 5;
  const int m    = lane & 15;                   // M (A) / N (B,D) index
  const int kh   = lane >> 4;                   // K half-select
  const int tokBase = blockIdx.y * TOK_PER_WG;
  const int outBase = blockIdx.x * 512 + wave * 64;

  // ---- Hoist every operand into registers (compiler can clause the loads) --
  v2f aop[4];                                   // t tile, 4 K-steps, shared by all NT
#pragma unroll
  for (int s = 0; s < 4; ++s) {
    const float* p = t + (size_t)(tokBase + m) * RANK + s * 4 + kh * 2;
    aop[s] = v2f{p[0], p[1]};
  }

  v2f  bop[NT][4];                              // B^T tiles: B^T[k][col] = B[col][k]
  v8f  acc[NT];                                 // C pre-loaded with bias[col]
#pragma unroll
  for (int n = 0; n < NT; ++n) {
    const int col = outBase + n * 16 + m;
#pragma unroll
    for (int s = 0; s < 4; ++s) {
      const float* p = Bw + (size_t)col * RANK + s * 4 + kh * 2;
      bop[n][s] = v2f{p[0], p[1]};
    }
    const float bv = bias[col];
#pragma unroll
    for (int v = 0; v < 8; ++v) acc[n][v] = bv;
  }

  // ---- 16 WMMAs, n fastest: consecutive ops are independent ----------------
#pragma unroll
  for (int s = 0; s < 4; ++s) {
#pragma unroll
    for (int n = 0; n < NT; ++n) {
      acc[n] = __builtin_amdgcn_wmma_f32_16x16x4_f32(
                   /*neg_a=*/false, aop[s], /*neg_b=*/false, bop[n][s],
                   /*c_mod=*/(short)0, acc[n], /*reuse_a=*/false, /*reuse_b=*/false);
    }
  }

  // ---- Repack D through LDS --------------------------------------------
#pragma unroll
  for (int n = 0; n < NT; ++n) {
#pragma unroll
    for (int v = 0; v < 8; ++v) {
      stile[wave][v + 8 * kh][n * 16 + m] = acc[n][v];   // row M, col in 64-wide tile
    }
  }
  __syncthreads();

  // Store: 16 rows x 64 cols per wave; 16 consecutive lanes cover one
  // 256 B row segment -> global_store_b128, fully-coalesced out stream.
#pragma unroll
  for (int i = 0; i < 8; ++i) {
    const int flat = i * 32 + lane;             // 0..255
    const int row  = flat >> 4;                 // 0..15
    const int ch   = flat & 15;                 // 16-byte chunk within 64-col row
    const float4 val = *(const float4*)&stile[wave][row][ch * 4];
    *(float4*)(out + (size_t)(tokBase + row) * OUT_F + outBase + ch * 4) = val;
  }
}

// ---------------------------------------------------------------------------
extern "C" void kernel_launch(void* const* d_in, const int* in_sizes, int n_in,
                              void* d_out, int out_size, void* d_ws, size_t ws_size,
                              hipStream_t stream) {
  const float* x    = (const float*)d_in[0];    // [4,2048,4096]
  const float* A    = (const float*)d_in[1];    // [16,4096]
  const float* Bw   = (const float*)d_in[2];    // [4096,16]
  const float* bias = (const float*)d_in[3];    // [4096]
  float* out = (float*)d_out;                   // [4,2048,4096]
  float* t   = (float*)d_ws;                    // [8192,16] = 512 KB scratch

  lora_xat<<<dim3(NTOK / TOK_PER_WG), dim3(256), 0, stream>>>(x, A, t);
  lora_tbt<<<dim3(OUT_F / 512, NTOK / TOK_PER_WG), dim3(256), 0, stream>>>(t, Bw, bias, out);
}